// E53SigmoidGateCell_45792941310469
// MI455X (gfx1250) — compile-verified
//
#include <hip/hip_runtime.h>
#include <math.h>

// ---------------------------------------------------------------------------
// Spectral-norm RNN for MI455X (gfx1250, wave32, WMMA f32 16x16x4)
//   s     = 0.999 / sigma(W)            (power iteration, 1 block)
//   h[t]  = s*(x_t @ W^T) + b + s*(h[t-1] @ W^T)    (GEMM + persistent scan)
//   out   = silu(h[1..T])
// d_out layout: output [T,B,D] floats, then h [T+1,B,D] floats.
// ---------------------------------------------------------------------------

typedef __attribute__((ext_vector_type(2))) float v2f;
typedef __attribute__((ext_vector_type(8))) float v8f;

#define T_STEPS 1024
#define BATCH   8
#define DIM     1024
#define BD      (BATCH * DIM)   // 8192
#define HPAD    (DIM + 4)       // LDS row stride: bank-spread + 16B aligned
#define SRAD    0.999f
#define EPSV    1e-8f

__device__ __forceinline__ v2f ldg2(const float* p) {
    return *reinterpret_cast<const v2f*>(p);
}

// D = A(16x4 f32) * B(4x16 f32) + C(16x16 f32)
__device__ __forceinline__ v8f wmma4(v2f a, v2f b, v8f c) {
    return __builtin_amdgcn_wmma_f32_16x16x4_f32(
        /*neg_a=*/false, a, /*neg_b=*/false, b,
        /*c_mod=*/(short)0, c, /*reuse_a=*/false, /*reuse_b=*/false);
}

// CDNA5 async copy: 16 bytes global -> LDS, tracked by ASYNCcnt.
// lds_off = wave-relative LDS byte address (low 32 bits of flat shared addr).
__device__ __forceinline__ void async_load_b128(unsigned lds_off, const float* gptr) {
    asm volatile("global_load_async_to_lds_b128 %0, %1, off scope:SCOPE_DEV"
                 :: "v"(lds_off), "v"(gptr) : "memory");
}

__device__ __forceinline__ void wait_asynccnt0() {
    asm volatile("s_wait_asynccnt 0x0" ::: "memory");
}

// ---------------------------------------------------------------------------
// Kernel 1: 3-step power iteration -> scale s; reset barrier counter; h[0]=h0
// ---------------------------------------------------------------------------
__global__ void spectral_scale_kernel(const float* __restrict__ W,
                                      const float* __restrict__ u,
                                      const float* __restrict__ h0,
                                      float* __restrict__ hbase,
                                      float* __restrict__ ws_scale,
                                      unsigned int* __restrict__ counter) {
    __shared__ float su[DIM];
    __shared__ float sv[DIM];
    __shared__ float sr[DIM];
    const int tid = threadIdx.x;

    su[tid] = u[tid];
    __syncthreads();

    float last_n = 0.f, last_n2 = 0.f;
    for (int it = 0; it < 3; ++it) {
        // v = W^T u ; v /= (||v|| + eps)
        float acc = 0.f;
        for (int i = 0; i < DIM; ++i) acc += W[i * DIM + tid] * su[i];
        sr[tid] = acc * acc;
        __syncthreads();
        for (int off = DIM / 2; off > 0; off >>= 1) {
            if (tid < off) sr[tid] += sr[tid + off];
            __syncthreads();
        }
        float vn = sqrtf(sr[0]);
        sv[tid] = acc / (vn + EPSV);
        __syncthreads();

        // u = W v ; u /= (||u|| + eps)
        float acc2 = 0.f;
        for (int j = 0; j < DIM; ++j) acc2 += W[tid * DIM + j] * sv[j];
        __syncthreads();                 // everyone done reading sr[0]/sv
        sr[tid] = acc2 * acc2;
        __syncthreads();
        for (int off = DIM / 2; off > 0; off >>= 1) {
            if (tid < off) sr[tid] += sr[tid + off];
            __syncthreads();
        }
        last_n2 = sr[0];
        last_n  = sqrtf(last_n2);
        su[tid] = acc2 / (last_n + EPSV);
        __syncthreads();
    }

    if (tid == 0) {
        // u_final = (Wv)/(||Wv||+eps)  =>  sigma = |u . (Wv)| = n^2/(n+eps)
        float sigma = last_n2 / (last_n + EPSV);
        ws_scale[0] = SRAD / (sigma + EPSV);
        *counter = 0u;                   // reset grid barrier (ws is poisoned)
    }

    // h[0] = h0
    for (int idx = tid; idx < BD; idx += blockDim.x) hbase[idx] = h0[idx];
}

// ---------------------------------------------------------------------------
// Kernel 2: h[1..T] = s * (X @ W^T) + b     (M=8192, N=1024, K=1024)
// 2x4 register blocking: each wave computes a 32x64 tile, 8 WMMAs / 6 loads.
// ---------------------------------------------------------------------------
__global__ void wx_gemm_kernel(const float* __restrict__ X,
                               const float* __restrict__ W,
                               const float* __restrict__ bias,
                               const float* __restrict__ ws_scale,
                               float* __restrict__ hbase) {
    const float s   = ws_scale[0];
    const int lane  = threadIdx.x & 31;
    const int wave  = threadIdx.x >> 5;             // 0..7
    const int m0    = blockIdx.y * 32;              // 256 m-supertiles
    const int n0    = (blockIdx.x * 8 + wave) * 64; // 16 n-supertiles
    const int fr    = lane & 15;                    // fragment row/col
    const int koff  = (lane >> 4) * 2;              // lanes 16..31: K+2,K+3

    const float* __restrict__ xrow0 = X + (size_t)(m0 + fr) * DIM + koff;
    const float* __restrict__ xrow1 = xrow0 + (size_t)16 * DIM;
    const float* __restrict__ wrow0 = W + (size_t)(n0 + fr) * DIM + koff;

    v8f acc[2][4];
#pragma unroll
    for (int i = 0; i < 2; ++i)
#pragma unroll
        for (int j = 0; j < 4; ++j)
            acc[i][j] = (v8f){0.f, 0.f, 0.f, 0.f, 0.f, 0.f, 0.f, 0.f};

#pragma unroll 2
    for (int k0 = 0; k0 < DIM; k0 += 4) {
        v2f a0 = ldg2(xrow0 + k0);                    // A rows m0..m0+15
        v2f a1 = ldg2(xrow1 + k0);                    // A rows m0+16..m0+31
        v2f b0 = ldg2(wrow0 + k0);                    // B[k][n] = W[n][k]
        v2f b1 = ldg2(wrow0 + (size_t)16 * DIM + k0);
        v2f b2 = ldg2(wrow0 + (size_t)32 * DIM + k0);
        v2f b3 = ldg2(wrow0 + (size_t)48 * DIM + k0);
        acc[0][0] = wmma4(a0, b0, acc[0][0]);
        acc[0][1] = wmma4(a0, b1, acc[0][1]);
        acc[0][2] = wmma4(a0, b2, acc[0][2]);
        acc[0][3] = wmma4(a0, b3, acc[0][3]);
        acc[1][0] = wmma4(a1, b0, acc[1][0]);
        acc[1][1] = wmma4(a1, b1, acc[1][1]);
        acc[1][2] = wmma4(a1, b2, acc[1][2]);
        acc[1][3] = wmma4(a1, b3, acc[1][3]);
    }

    const int mtop = (lane >> 4) * 8;                 // C rows r / r+8
#pragma unroll
    for (int j = 0; j < 4; ++j) {
        const int nc = n0 + j * 16 + fr;
        const float bn = bias[nc];
#pragma unroll
        for (int i = 0; i < 2; ++i) {
#pragma unroll
            for (int r = 0; r < 8; ++r) {
                int m = m0 + i * 16 + mtop + r;       // global row = t*B+batch
                hbase[(size_t)BD + (size_t)m * DIM + nc] = s * acc[i][j][r] + bn;
            }
        }
    }
}

// ---------------------------------------------------------------------------
// Kernel 3: persistent scan. 64 blocks (one 16-col tile each) x 4 waves.
// h[t-1] staged to LDS via GLOBAL_LOAD_ASYNC_TO_LDS_B128 (ASYNCcnt), then
// WMMA partials over K-quarters, LDS reduce, h[t] += ..., out = silu(h[t]).
// Grid barrier per step via device-scope atomic + s_sleep spin.
// ---------------------------------------------------------------------------
__global__ void recurrence_kernel(const float* __restrict__ W,
                                  const float* __restrict__ ws_scale,
                                  float* __restrict__ hbase,
                                  float* __restrict__ out,
                                  unsigned int* __restrict__ counter) {
    const float s   = ws_scale[0];
    const int tid   = threadIdx.x;
    const int lane  = tid & 31;
    const int wave  = tid >> 5;           // 0..3, each owns K/4 = 256
    const int n0    = blockIdx.x * 16;
    const int ncol  = n0 + (lane & 15);
    const int m     = lane & 15;          // A row (valid if < BATCH)
    const int koff  = (lane >> 4) * 2;
    const int kbase = wave * 256;

    __shared__ float hsh[BATCH][HPAD];    // staged h[t-1], padded rows
    __shared__ float red[4][32][8];

    const unsigned hsh_base = (unsigned)(uintptr_t)&hsh[0][0];
    const float* __restrict__ wrow = W + (size_t)ncol * DIM + koff;
    const float* __restrict__ arow = &hsh[0][0] + (size_t)m * HPAD + koff;

    for (int t = 1; t <= T_STEPS; ++t) {
        const float* __restrict__ hp = hbase + (size_t)(t - 1) * BD;

        // ---- async-stage h[t-1] (8 x 1024 f32 = 2048 x b128) into LDS ----
#pragma unroll
        for (int j = 0; j < 16; ++j) {
            int c  = j * 128 + tid;                  // chunk id 0..2047
            int mm = c >> 8;                         // row 0..7 (256 chunks/row)
            int kc = (c & 255) * 4;                  // starting float in row
            unsigned ldso = hsh_base + (unsigned)((mm * HPAD + kc) * 4);
            async_load_b128(ldso, hp + (size_t)mm * DIM + kc);
        }
        wait_asynccnt0();
        __syncthreads();

        // ---- WMMA partial over this wave's K range ----
        v8f acc = {0.f, 0.f, 0.f, 0.f, 0.f, 0.f, 0.f, 0.f};
#pragma unroll 4
        for (int kk = kbase; kk < kbase + 256; kk += 4) {
            v2f a = {0.f, 0.f};
            if (m < BATCH) a = *reinterpret_cast<const v2f*>(arow + kk);
            v2f b = ldg2(wrow + kk);                 // scale folded in later
            acc = wmma4(a, b, acc);
        }

#pragma unroll
        for (int r = 0; r < 8; ++r) red[wave][lane][r] = acc[r];
        __syncthreads();

        if (wave == 0 && lane < 16) {
            // lanes 0..15 hold C rows r = 0..7 (== batch index), col = ncol
#pragma unroll
            for (int r = 0; r < 8; ++r) {
                float v = s * (red[0][lane][r] + red[1][lane][r] +
                               red[2][lane][r] + red[3][lane][r]);
                size_t idx = (size_t)t * BD + (size_t)r * DIM + ncol;
                float hv = hbase[idx] + v;           // add to s*(x@W^T)+b
                hbase[idx] = hv;
                float sg = 1.f / (1.f + __expf(-hv));
                out[(size_t)(t - 1) * BD + (size_t)r * DIM + ncol] = hv * sg;
            }
        }

        // ---- grid-wide barrier: release h[t], wait, acquire ----
        __threadfence();
        __syncthreads();
        if (tid == 0) {
            atomicAdd(counter, 1u);
            const unsigned int target = (unsigned int)gridDim.x * (unsigned int)t;
            while (__hip_atomic_load(counter, __ATOMIC_ACQUIRE,
                                     __HIP_MEMORY_SCOPE_AGENT) < target) {
                __builtin_amdgcn_s_sleep(1);
            }
        }
        __syncthreads();
        __threadfence();
    }
}

// ---------------------------------------------------------------------------
extern "C" void kernel_launch(void* const* d_in, const int* in_sizes, int n_in,
                              void* d_out, int out_size, void* d_ws, size_t ws_size,
                              hipStream_t stream) {
    const float* x  = (const float*)d_in[0];   // [T,B,D]
    const float* h0 = (const float*)d_in[1];   // [B,D]
    const float* W  = (const float*)d_in[2];   // [D,D]
    const float* b  = (const float*)d_in[3];   // [D]
    const float* u  = (const float*)d_in[4];   // [D]

    float* out   = (float*)d_out;                       // [T,B,D]
    float* hbase = out + (size_t)T_STEPS * BD;          // [T+1,B,D]

    float*        ws_scale = (float*)d_ws;
    unsigned int* counter  = (unsigned int*)((char*)d_ws + 64);

    spectral_scale_kernel<<<1, DIM, 0, stream>>>(W, u, h0, hbase, ws_scale, counter);
    wx_gemm_kernel<<<dim3(2, 256), 256, 0, stream>>>(x, W, b, ws_scale, hbase);
    recurrence_kernel<<<64, 128, 0, stream>>>(W, ws_scale, hbase, out, counter);
}